// StubAttention_33260226740637
// MI455X (gfx1250) — compile-verified
//
#include <hip/hip_runtime.h>
#include <hip/hip_bf16.h>
#include <math.h>

#define HIDDEN    2048
#define HEAD_DIM  128
#define NUM_HEADS 16
#define NUM_KV    8
#define MAX_SEQ   2048
#define LAYER_IDX 5
#define ATT_SCALE 0.08838834764831845f   // 1/sqrt(128)

typedef __attribute__((ext_vector_type(2))) float v2f;
typedef __attribute__((ext_vector_type(8))) float v8f;

// ---------------------------------------------------------------------------
// Kernel 1: fused Q/K/V projection.  4096 output rows (2048 q, 1024 k, 1024 v).
// One block = 16 rows, 8 waves split the K=2048 reduction (256 each, 64 WMMAs),
// partial dot products combined through LDS (fixed order -> deterministic).
// A tile  : W[m0+ (lane&15)][k + (lane>>4)*2 .. +1]    (16x4 f32, ISA layout)
// B tile  : x broadcast into all 16 columns (column 0 is the one we read back)
// ---------------------------------------------------------------------------
__global__ __launch_bounds__(256) void qkv_proj_kernel(
    const float* __restrict__ x,
    const float* __restrict__ qw, const float* __restrict__ kw,
    const float* __restrict__ vw, float* __restrict__ qkv)
{
    __shared__ float xs[HIDDEN];
    __shared__ float partial[8][16];

    const int tid = threadIdx.x;
    for (int i = tid; i < HIDDEN; i += 256) xs[i] = x[i];
    __syncthreads();

    const int rowBase = blockIdx.x * 16;          // 0..4095, 16-aligned
    const float* W;  int localRow;
    if (rowBase < 2048)      { W = qw; localRow = rowBase;        }
    else if (rowBase < 3072) { W = kw; localRow = rowBase - 2048; }
    else                     { W = vw; localRow = rowBase - 3072; }

    const int wave = tid >> 5;
    const int lane = tid & 31;
    const int m    = lane & 15;
    const int koff = (lane >> 4) * 2;             // 0 or 2

    const float* wrow = W + (size_t)(localRow + m) * HIDDEN + koff;
    const int k0 = wave * 256;

    v8f c = {};
    for (int k = k0; k < k0 + 256; k += 4) {
        v2f a = *(const v2f*)(wrow + k);
        v2f b = *(const v2f*)(&xs[k + koff]);     // broadcast across columns
        c = __builtin_amdgcn_wmma_f32_16x16x4_f32(false, a, false, b,
                                                  (short)0, c, false, false);
    }

    // column 0 of D: rows 0-7 live in lane 0, rows 8-15 in lane 16
    if (lane == 0) {
#pragma unroll
        for (int i = 0; i < 8; ++i) partial[wave][i] = c[i];
    } else if (lane == 16) {
#pragma unroll
        for (int i = 0; i < 8; ++i) partial[wave][8 + i] = c[i];
    }
    __syncthreads();

    if (tid < 16) {
        float s = 0.f;
#pragma unroll
        for (int w = 0; w < 8; ++w) s += partial[w][tid];
        qkv[rowBase + tid] = s;
    }
}

// ---------------------------------------------------------------------------
// Kernel 2: attention for one kv head per block (8 blocks, 1024 thr = 32 waves)
// Phase 1: scores = K_tile(16 seq x 128) . q^T  via WMMA, B cols 0/1 = 2 heads
// Softmax: fixed-order tree reduction (deterministic)
// Phase 2: attn = P(2 x S) . V  via WMMA; dim-tile x seq-quarter split
// current_pos row redirected to freshly-computed k/v (input cache not mutated)
// ---------------------------------------------------------------------------
__global__ __launch_bounds__(1024) void attn_kernel(
    const float* __restrict__ kvc, const float* __restrict__ qkv,
    const int* __restrict__ posp, float* __restrict__ attn_out)
{
    __shared__ float q_s[16][HEAD_DIM];      // rows 2..15 zero-padded
    __shared__ float sc[2][MAX_SEQ];         // scores -> unnormalized probs
    __shared__ float rbuf[2][512];
    __shared__ float hsum[2];
    __shared__ float pavt[4][2][HEAD_DIM];   // [seq quarter][head][dim]

    const int g   = blockIdx.x;
    const int tid = threadIdx.x;
    const int pos = *posp;

    for (int i = tid; i < 16 * HEAD_DIM; i += 1024) {
        int hh = i / HEAD_DIM, d = i % HEAD_DIM;
        q_s[hh][d] = (hh < 2) ? qkv[(2 * g + hh) * HEAD_DIM + d] : 0.f;
    }
    __syncthreads();

    const float* kbase = kvc + ((size_t)(2 * LAYER_IDX)     * NUM_KV + g) * MAX_SEQ * HEAD_DIM;
    const float* vbase = kvc + ((size_t)(2 * LAYER_IDX + 1) * NUM_KV + g) * MAX_SEQ * HEAD_DIM;
    const float* knew  = qkv + 2048 + g * HEAD_DIM;
    const float* vnew  = qkv + 3072 + g * HEAD_DIM;

    const int wave  = tid >> 5;
    const int lane  = tid & 31;
    const int m     = lane & 15;
    const int koff2 = (lane >> 4) * 2;

    // ---- Phase 1: scores (128 seq tiles over 32 waves, 4 each) ----
    for (int i = 0; i < 4; ++i) {
        const int s0   = (wave * 4 + i) * 16;
        const int srow = s0 + m;
        const float* arow = (srow == pos) ? knew
                                          : (kbase + (size_t)srow * HEAD_DIM);
        v8f c = {};
        for (int k = 0; k < HEAD_DIM; k += 4) {
            v2f a = *(const v2f*)(arow + k + koff2);
            v2f b = *(const v2f*)(&q_s[m][k + koff2]);   // cols>=2 are zeros
            c = __builtin_amdgcn_wmma_f32_16x16x4_f32(false, a, false, b,
                                                      (short)0, c, false, false);
        }
        const int n = m;                    // column = q-head-in-group
        if (n < 2) {
            const int mb = (lane < 16) ? 0 : 8;
#pragma unroll
            for (int r = 0; r < 8; ++r)
                sc[n][s0 + mb + r] = c[r] * ATT_SCALE;
        }
    }
    __syncthreads();

    // ---- Softmax (512 threads per head) ----
    const int hh = tid >> 9;
    const int t2 = tid & 511;
    float mx = -1e30f;
    for (int s = t2; s < MAX_SEQ; s += 512) mx = fmaxf(mx, sc[hh][s]);
    rbuf[hh][t2] = mx;
    __syncthreads();
    for (int off = 256; off > 0; off >>= 1) {
        if (t2 < off) rbuf[hh][t2] = fmaxf(rbuf[hh][t2], rbuf[hh][t2 + off]);
        __syncthreads();
    }
    const float themax = rbuf[hh][0];
    __syncthreads();
    float sum = 0.f;
    for (int s = t2; s < MAX_SEQ; s += 512) {
        float p = __expf(sc[hh][s] - themax);
        sc[hh][s] = p;
        sum += p;
    }
    rbuf[hh][t2] = sum;
    __syncthreads();
    for (int off = 256; off > 0; off >>= 1) {
        if (t2 < off) rbuf[hh][t2] += rbuf[hh][t2 + off];
        __syncthreads();
    }
    if (t2 == 0) hsum[hh] = rbuf[hh][0];
    __syncthreads();

    // ---- Phase 2: attn = P . V  (8 dim tiles x 4 seq quarters = 32 waves) ----
    const int dt = wave & 7, quarter = wave >> 3;
    const int d0 = dt * 16;
    const int sbeg = quarter * 512;
    v8f c = {};
    for (int s = sbeg; s < sbeg + 512; s += 4) {
        float p0 = sc[m & 1][s + koff2];
        float p1 = sc[m & 1][s + koff2 + 1];
        v2f a;  a[0] = (m < 2) ? p0 : 0.f;  a[1] = (m < 2) ? p1 : 0.f;
        const int sb = s + koff2;
        const float* vr0 = (sb     == pos) ? vnew : (vbase + (size_t)sb * HEAD_DIM);
        const float* vr1 = (sb + 1 == pos) ? vnew : (vbase + (size_t)(sb + 1) * HEAD_DIM);
        v2f b;  b[0] = vr0[d0 + m];  b[1] = vr1[d0 + m];
        c = __builtin_amdgcn_wmma_f32_16x16x4_f32(false, a, false, b,
                                                  (short)0, c, false, false);
    }
    if (lane < 16) {                        // D rows 0/1 = heads, cols = dims
        pavt[quarter][0][d0 + lane] = c[0];
        pavt[quarter][1][d0 + lane] = c[1];
    }
    __syncthreads();

    if (tid < 256) {
        const int h2 = tid >> 7, d = tid & 127;
        float val = (pavt[0][h2][d] + pavt[1][h2][d] +
                     pavt[2][h2][d] + pavt[3][h2][d]) / hsum[h2];
        attn_out[(2 * g + h2) * HEAD_DIM + d] = val;
    }
}

// ---------------------------------------------------------------------------
// Kernel 3: out = o_w @ attn  (2048x2048 GEMV, same scheme as kernel 1)
// ---------------------------------------------------------------------------
__global__ __launch_bounds__(256) void out_proj_kernel(
    const float* __restrict__ ow, const float* __restrict__ attn,
    float* __restrict__ out)
{
    __shared__ float xs[HIDDEN];
    __shared__ float partial[8][16];

    const int tid = threadIdx.x;
    for (int i = tid; i < HIDDEN; i += 256) xs[i] = attn[i];
    __syncthreads();

    const int rowBase = blockIdx.x * 16;
    const int wave = tid >> 5;
    const int lane = tid & 31;
    const int m    = lane & 15;
    const int koff = (lane >> 4) * 2;

    const float* wrow = ow + (size_t)(rowBase + m) * HIDDEN + koff;
    const int k0 = wave * 256;

    v8f c = {};
    for (int k = k0; k < k0 + 256; k += 4) {
        v2f a = *(const v2f*)(wrow + k);
        v2f b = *(const v2f*)(&xs[k + koff]);
        c = __builtin_amdgcn_wmma_f32_16x16x4_f32(false, a, false, b,
                                                  (short)0, c, false, false);
    }
    if (lane == 0) {
#pragma unroll
        for (int i = 0; i < 8; ++i) partial[wave][i] = c[i];
    } else if (lane == 16) {
#pragma unroll
        for (int i = 0; i < 8; ++i) partial[wave][8 + i] = c[i];
    }
    __syncthreads();
    if (tid < 16) {
        float s = 0.f;
#pragma unroll
        for (int w = 0; w < 8; ++w) s += partial[w][tid];
        out[rowBase + tid] = s;
    }
}

// ---------------------------------------------------------------------------
extern "C" void kernel_launch(void* const* d_in, const int* in_sizes, int n_in,
                              void* d_out, int out_size, void* d_ws, size_t ws_size,
                              hipStream_t stream)
{
    (void)in_sizes; (void)n_in; (void)out_size; (void)ws_size;
    const float* x   = (const float*)d_in[0];
    const float* kvc = (const float*)d_in[1];
    const float* qw  = (const float*)d_in[2];
    const float* kw  = (const float*)d_in[3];
    const float* vw  = (const float*)d_in[4];
    const float* ow  = (const float*)d_in[5];
    const int*   pos = (const int*)  d_in[6];
    float* out = (float*)d_out;
    float* ws  = (float*)d_ws;
    // ws layout (floats): [0,2048) q | [2048,3072) k | [3072,4096) v | [4096,6144) attn

    qkv_proj_kernel<<<256, 256, 0, stream>>>(x, qw, kw, vw, ws);
    attn_kernel   <<<  8, 1024, 0, stream>>>(kvc, ws, pos, ws + 4096);
    out_proj_kernel<<<128, 256, 0, stream>>>(ow, ws + 4096, out);
}